// MSHTrans_4681514353070
// MI455X (gfx1250) — compile-verified
//
#include <hip/hip_runtime.h>
#include <hip/hip_bf16.h>

typedef float v2f __attribute__((ext_vector_type(2)));
typedef float v8f __attribute__((ext_vector_type(8)));

#define TPB 256
static inline dim3 g1(long long n) { return dim3((unsigned)((n + TPB - 1) / TPB)); }

// ---------------------------------------------------------------------------
// WMMA f32 GEMM:  C[M,N] = op(A) @ W[K,N] + bias
//   MODE 0: op(A)=A1   MODE 1: op(A)=A1-A2   MODE 2: op(A)=A1+A2
// One wave -> one 16x16 C tile via V_WMMA_F32_16X16X4_F32, K stepped by 4.
// W tile (Kx16) is staged once per block into LDS, pre-swizzled into
// (k, k+1) pairs so each B fragment is one conflict-free ds_load_b64.
// Fragment layouts per CDNA5 ISA 7.12.2:
//   A 16x4 : lane&15 = M row, VGPR v holds K = v + 2*(lane>>4)
//   B 4x16 : lane&15 = N col, VGPR v holds K = v + 2*(lane>>4)
//   C 16x16: VGPR r -> M = r + 8*(lane>>4), N = lane&15
// ---------------------------------------------------------------------------
template <int MODE>
__global__ void k_gemm_wmma(const float* __restrict__ A1, const float* __restrict__ A2,
                            const float* __restrict__ W, const float* __restrict__ bias,
                            float* __restrict__ C, int M, int K, int N)
{
    __shared__ float ldsW[2048];               // max K=128 -> K*16 floats (8 KB)
    int tid  = threadIdx.x;
    int wave = tid >> 5;
    int lane = tid & 31;
    int tile_m = blockIdx.y * 8 + wave;
    int n0 = blockIdx.x << 4;

    // cooperative stage of W[0:K, n0:n0+16] into pair-swizzled LDS
    for (int i = tid; i < K * 16; i += TPB) {
        int n = i & 15;
        int k = i >> 4;
        int kk = k >> 2, q = k & 3;
        // pair slot s = kk*2 + (q>>1); element = q&1
        ldsW[((kk * 2 + (q >> 1)) * 16 + n) * 2 + (q & 1)] = W[k * N + n0 + n];
    }
    __syncthreads();
    if (tile_m * 16 >= M) return;              // wave-uniform; after barrier

    int m0  = tile_m << 4;
    int r15 = lane & 15;
    int kh  = lane >> 4;
    const float* ap  = A1 + (m0 + r15) * K + 2 * kh;
    const float* ap2 = MODE ? (A2 + (m0 + r15) * K + 2 * kh) : (const float*)0;
    const float* wp  = ldsW + kh * 32 + r15 * 2;
    int ncol = n0 + r15;

    v8f acc = {};
    for (int kk = 0; kk < (K >> 2); ++kk) {
        v2f a = *(const v2f*)ap;  ap += 4;
        if (MODE == 1) { v2f a2 = *(const v2f*)ap2; ap2 += 4; a.x -= a2.x; a.y -= a2.y; }
        if (MODE == 2) { v2f a2 = *(const v2f*)ap2; ap2 += 4; a.x += a2.x; a.y += a2.y; }
        v2f b = *(const v2f*)wp;  wp += 64;
        acc = __builtin_amdgcn_wmma_f32_16x16x4_f32(false, a, false, b,
                                                    (short)0, acc, false, false);
    }
    float bv = bias ? bias[ncol] : 0.0f;
    float* cp = C + (m0 + 8 * kh) * N + ncol;
#pragma unroll
    for (int r = 0; r < 8; ++r) cp[r * N] = acc[r] + bv;
}

// ---- positional encoding helper -------------------------------------------
__device__ __forceinline__ float posenc(int t, int c, int d)
{
    int k = c & ~1;
    float dv = __expf(-__logf(10000.0f) * (float)k / (float)d);
    float a = (float)t * dv;
    return (c & 1) ? __cosf(a) : __sinf(a);
}

// ---- strided conv pooling (P==4): ELU(sum_{p,f} in[b,t*4+p,f]*K[o,f,p]+Kb[o])
__global__ void k_pool_conv(const float* __restrict__ in, const float* __restrict__ Kw,
                            const float* __restrict__ Kb, float* __restrict__ out,
                            int Bn, int Tout, int F)
{
    int idx = blockIdx.x * TPB + threadIdx.x;
    int total = Bn * Tout * F;
    if (idx >= total) return;
    int o = idx % F;
    int t = (idx / F) % Tout;
    int b = idx / (F * Tout);
    const float* ip = in + (long long)((b * Tout + t) * 4) * F;
    float s = Kb[o];
    for (int f = 0; f < F; ++f) {
        float4 kw = *(const float4*)(Kw + (o * F + f) * 4);   // all 4 taps of (o,f)
        s += ip[f] * kw.x + ip[F + f] * kw.y + ip[2 * F + f] * kw.z + ip[3 * F + f] * kw.w;
    }
    out[idx] = s > 0.0f ? s : (__expf(s) - 1.0f);
}

// ---- layer norm over last dim F -------------------------------------------
__global__ void k_layernorm(const float* __restrict__ in, const float* __restrict__ g,
                            const float* __restrict__ b, float* __restrict__ out,
                            int rows, int F)
{
    int r = blockIdx.x * TPB + threadIdx.x;
    if (r >= rows) return;
    const float* p = in + (long long)r * F;
    float mu = 0.f;
    for (int i = 0; i < F; ++i) mu += p[i];
    mu /= F;
    float v = 0.f;
    for (int i = 0; i < F; ++i) { float d = p[i] - mu; v += d * d; }
    v /= F;
    float inv = rsqrtf(v + 1e-5f);
    float* o = out + (long long)r * F;
    for (int i = 0; i < F; ++i) o[i] = (p[i] - mu) * inv * g[i] + b[i];
}

// ---- e = concat(seq_enc, tail) + posenc ------------------------------------
__global__ void k_build_e(const float* __restrict__ se, const float* __restrict__ x,
                          float* __restrict__ e, int Bn, int N, int F, int stride, int T)
{
    int idx = blockIdx.x * TPB + threadIdx.x;
    int D2 = 2 * F;
    int total = Bn * N * D2;
    if (idx >= total) return;
    int c = idx % D2;
    int n = (idx / D2) % N;
    int b = idx / (D2 * N);
    float v = (c < F) ? se[(long long)(b * N + n) * F + c]
                      : x[(long long)(b * T + n * stride) * F + (c - F)];
    e[idx] = v + posenc(n, c, D2);
}

__global__ void k_fill4(float4* __restrict__ p, int n4)
{
    int i = blockIdx.x * TPB + threadIdx.x;
    if (i < n4) p[i] = make_float4(0.f, 0.f, 0.f, 0.f);
}

__global__ void k_avg_head(const float* __restrict__ in, float* __restrict__ out, int Hn, int sz)
{
    int j = blockIdx.x * TPB + threadIdx.x;
    if (j >= sz) return;
    float s = 0.f;
    for (int h = 0; h < Hn; ++h) s += in[(long long)h * sz + j];
    out[j] = s / (float)Hn;
}

__global__ void k_edeg(const int* __restrict__ edge, int nnz, float* __restrict__ deg)
{
    int i = blockIdx.x * TPB + threadIdx.x;
    if (i >= nnz) return;
    atomicAdd(&deg[edge[i]], 1.0f);
}

// dst(E,B,Dw) += src(B,N,Dw) scattered by (node -> edge)
__global__ void k_scatter(float* __restrict__ dst, const float* __restrict__ src,
                          const int* __restrict__ node, const int* __restrict__ edge,
                          int nnz, int Bn, int Dw, int N)
{
    int idx = blockIdx.x * TPB + threadIdx.x;
    int total = nnz * Bn * Dw;
    if (idx >= total) return;
    int d = idx % Dw;
    int b = (idx / Dw) % Bn;
    int i = idx / (Dw * Bn);
    atomicAdd(&dst[(long long)(edge[i] * Bn + b) * Dw + d],
              src[(long long)(b * N + node[i]) * Dw + d]);
}

// He = He/max(edeg,1) + Ee  (in place, float4)
__global__ void k_msg4(float4* __restrict__ He, const float4* __restrict__ Ee,
                       const float* __restrict__ deg, int E, int Bn, int Dw)
{
    int q = Bn * Dw / 4;
    int idx = blockIdx.x * TPB + threadIdx.x;
    if (idx >= E * q) return;
    float dg = deg[idx / q];
    dg = dg < 1.0f ? 1.0f : dg;
    float inv = 1.0f / dg;
    float4 h = He[idx], e = Ee[idx];
    He[idx] = make_float4(h.x * inv + e.x, h.y * inv + e.y, h.z * inv + e.z, h.w * inv + e.w);
}

// out[b,n,:] = mean_j msg[edge[3n+j],b,:] + bbar (+ resid), float4 over d
__global__ void k_gather4(const float* __restrict__ msg, const int* __restrict__ edge,
                          const float* __restrict__ bbar, const float* __restrict__ resid,
                          float* __restrict__ out, int Bn, int N, int Dw)
{
    int Q = Dw >> 2;
    int idx = blockIdx.x * TPB + threadIdx.x;
    int total = Bn * N * Q;
    if (idx >= total) return;
    int d4 = idx % Q;
    int n = (idx / Q) % N;
    int b = idx / (Q * N);
    int e0 = edge[3 * n], e1 = edge[3 * n + 1], e2 = edge[3 * n + 2];
    float4 m0 = *(const float4*)(msg + (long long)(e0 * Bn + b) * Dw + 4 * d4);
    float4 m1 = *(const float4*)(msg + (long long)(e1 * Bn + b) * Dw + 4 * d4);
    float4 m2 = *(const float4*)(msg + (long long)(e2 * Bn + b) * Dw + 4 * d4);
    float4 bb = *(const float4*)(bbar + 4 * d4);
    const float third = 1.0f / 3.0f;
    float4 v = make_float4((m0.x + m1.x + m2.x) * third + bb.x,
                           (m0.y + m1.y + m2.y) * third + bb.y,
                           (m0.z + m1.z + m2.z) * third + bb.z,
                           (m0.w + m1.w + m2.w) * third + bb.w);
    if (resid) {
        float4 r = *(const float4*)(resid + (long long)idx * 4);
        v.x += r.x; v.y += r.y; v.z += r.z; v.w += r.w;
    }
    *(float4*)(out + (long long)idx * 4) = v;
}

// 25-tap moving average with edge padding, float4 over channels
__global__ void k_movavg4(const float* __restrict__ in, float* __restrict__ out,
                          int Bn, int T, int C)
{
    int Q = C >> 2;
    int idx = blockIdx.x * TPB + threadIdx.x;
    int total = Bn * T * Q;
    if (idx >= total) return;
    int c4 = idx % Q;
    int t = (idx / Q) % T;
    int b = idx / (Q * T);
    float4 s = make_float4(0.f, 0.f, 0.f, 0.f);
    for (int j = -12; j <= 12; ++j) {
        int tt = t + j;
        tt = tt < 0 ? 0 : (tt >= T ? T - 1 : tt);
        float4 v = *(const float4*)(in + (long long)(b * T + tt) * C + 4 * c4);
        s.x += v.x; s.y += v.y; s.z += v.z; s.w += v.w;
    }
    const float k = 1.0f / 25.0f;
    *(float4*)(out + (long long)idx * 4) =
        make_float4(s.x * k, s.y * k, s.z * k, s.w * k);
}

__device__ __forceinline__ float sigm(float v) { return 1.0f / (1.0f + __expf(-v)); }

__global__ void k_stcomb4(const float4* __restrict__ G, const float4* __restrict__ S,
                          const float4* __restrict__ Tr, float4* __restrict__ out,
                          int n4, int final_sig)
{
    int idx = blockIdx.x * TPB + threadIdx.x;
    if (idx >= n4) return;
    float4 g = G[idx], s = S[idx], t = Tr[idx];
    float4 v = make_float4(sigm(g.x) * s.x + (1.f - sigm(g.x)) * t.x,
                           sigm(g.y) * s.y + (1.f - sigm(g.y)) * t.y,
                           sigm(g.z) * s.z + (1.f - sigm(g.z)) * t.z,
                           sigm(g.w) * s.w + (1.f - sigm(g.w)) * t.w);
    if (final_sig) v = make_float4(sigm(v.x), sigm(v.y), sigm(v.z), sigm(v.w));
    out[idx] = v;
}

__global__ void k_upsum4(float* __restrict__ st0, const float* __restrict__ st1,
                         const float* __restrict__ st2, int Bn, int T, int Dw)
{
    int Q = Dw >> 2;
    int idx = blockIdx.x * TPB + threadIdx.x;
    int total = Bn * T * Q;
    if (idx >= total) return;
    int d4 = idx % Q;
    int t = (idx / Q) % T;
    int b = idx / (Q * T);
    float4 a = *(const float4*)(st0 + (long long)idx * 4);
    float4 u1 = *(const float4*)(st1 + (long long)(b * (T / 4) + t / 4) * Dw + 4 * d4);
    float4 u2 = *(const float4*)(st2 + (long long)(b * (T / 16) + t / 16) * Dw + 4 * d4);
    *(float4*)(st0 + (long long)idx * 4) =
        make_float4(a.x + u1.x + u2.x, a.y + u1.y + u2.y,
                    a.z + u1.z + u2.z, a.w + u1.w + u2.w);
}

// decoder input: concat(x - trend1, fused_logits) + posenc
__global__ void k_dec_inp(const float* __restrict__ x, const float* __restrict__ trend1,
                          const float* __restrict__ flog, float* __restrict__ out,
                          int Bn, int T, int F)
{
    int idx = blockIdx.x * TPB + threadIdx.x;
    int D2 = 2 * F;
    int total = Bn * T * D2;
    if (idx >= total) return;
    int c = idx % D2;
    int t = (idx / D2) % T;
    int b = idx / (D2 * T);
    long long base = (long long)(b * T + t) * F;
    float v = (c < F) ? (x[base + c] - trend1[base + c]) : flog[base + (c - F)];
    out[idx] = v + posenc(t, c, D2);
}

// ---------------------------------------------------------------------------
extern "C" void kernel_launch(void* const* d_in, const int* in_sizes, int n_in,
                              void* d_out, int out_size, void* d_ws, size_t ws_size,
                              hipStream_t stream)
{
    (void)n_in; (void)out_size; (void)ws_size;
    const float *x, *Wd, *bd, *K0, *K1, *Kb0, *Kb1, *Wu, *bu, *lng, *lnb;
    const float *encW, *encb, *encWe;
    const float *fWg, *fbg, *fWs, *fbs, *fWt, *fbt;
    const float *msW, *msb, *dW, *db, *dWe;
    const float *dfWg, *dfbg, *dfWs, *dfbs, *dfWt, *dfbt;
    const int *hgp[3], *fhg;
    const float** F32 = (const float**)d_in;
    if (in_sizes[0] == 8 * 4096 * 64) {
        // recursive dict-insertion-order flattening
        x = F32[0];
        Wd = F32[1]; bd = F32[2]; K0 = F32[3]; K1 = F32[4]; Kb0 = F32[5]; Kb1 = F32[6];
        Wu = F32[7]; bu = F32[8]; lng = F32[9]; lnb = F32[10];
        encW = F32[11]; encb = F32[12]; encWe = F32[13];
        fWg = F32[14]; fbg = F32[15]; fWs = F32[16]; fbs = F32[17]; fWt = F32[18]; fbt = F32[19];
        msW = F32[20]; msb = F32[21];
        dW = F32[22]; db = F32[23]; dWe = F32[24];
        dfWg = F32[25]; dfbg = F32[26]; dfWs = F32[27]; dfbs = F32[28]; dfWt = F32[29]; dfbt = F32[30];
        hgp[0] = (const int*)d_in[31]; hgp[1] = (const int*)d_in[32];
        hgp[2] = (const int*)d_in[33]; fhg = (const int*)d_in[34];
    } else {
        // jax tree_flatten (alphabetically sorted keys at every level)
        fhg = (const int*)d_in[0];
        hgp[0] = (const int*)d_in[1]; hgp[1] = (const int*)d_in[2]; hgp[2] = (const int*)d_in[3];
        K0 = F32[4]; K1 = F32[5]; Kb0 = F32[6]; Kb1 = F32[7];
        Wd = F32[8]; Wu = F32[9]; lnb = F32[10]; bd = F32[11]; bu = F32[12]; lng = F32[13];
        dW = F32[14]; dWe = F32[15]; db = F32[16];
        encW = F32[17]; encWe = F32[18]; encb = F32[19];
        fWg = F32[20]; fWs = F32[21]; fWt = F32[22]; fbg = F32[23]; fbs = F32[24]; fbt = F32[25];
        dfWg = F32[26]; dfWs = F32[27]; dfWt = F32[28]; dfbg = F32[29]; dfbs = F32[30]; dfbt = F32[31];
        msW = F32[32]; msb = F32[33];
        x = F32[34];
    }

    // ---- workspace arena (floats) ----
    float* ws = (float*)d_ws;
    size_t off = 0;
    auto alloc = [&](size_t n) { float* p = ws + off; off += n; return p; };
    float* Hb  = alloc(2097152);   // h = x@Wd (B,4096,64)
    float* S1  = alloc(524288);    // scale1 (B,1024,64)
    float* S2  = alloc(131072);    // scale2 (B,256,64)
    float* SE  = alloc(2097152);   // seq_enc_i / fused_logits
    float* Eb  = alloc(4194304);   // e_i / decoder inp (B,N,128)
    float* EF  = alloc(1048576);   // ef (E,B,D)
    float* Yb  = alloc(4194304);   // Y = e@Wbar
    float* HE  = alloc(1048576);   // He -> msg
    float* EE  = alloc(1048576);   // Ee
    float* MH  = alloc(4194304);   // mh
    float* TR  = alloc(4194304);   // trend / dec trend1
    float* TA  = alloc(4194304);   // gate proj / dec trend2
    float* TB  = alloc(4194304);   // season proj / dec gate
    float* TC  = alloc(4194304);   // trend proj / dec season proj
    float* ST0 = alloc(4194304);
    float* ST1 = alloc(1048576);
    float* ST2 = alloc(262144);
    float* WB  = alloc(131072);    // head-averaged weights
    float* DEG = alloc(4096);

    float* encWbar  = WB;                // 3 x 128x128
    float* encWebar = WB + 49152;        // 3 x 128x128
    float* encbbar  = WB + 98304;        // 3 x 128
    float* decWbar  = WB + 98688;        // 128x64
    float* decWebar = WB + 106880;       // 64x64
    float* decbbar  = WB + 110976;       // 64

    auto gemm = [&](const float* A1, const float* A2, const float* W, const float* bias,
                    float* C, int M, int K, int N, int mode) {
        dim3 grid(N / 16, (M / 16 + 7) / 8);
        if (mode == 0)
            k_gemm_wmma<0><<<grid, 256, 0, stream>>>(A1, A2, W, bias, C, M, K, N);
        else if (mode == 1)
            k_gemm_wmma<1><<<grid, 256, 0, stream>>>(A1, A2, W, bias, C, M, K, N);
        else
            k_gemm_wmma<2><<<grid, 256, 0, stream>>>(A1, A2, W, bias, C, M, K, N);
    };

    // ---- fold the 4 heads into averaged weights (exact: hyconv is linear in h) ----
    for (int i = 0; i < 3; ++i) {
        k_avg_head<<<g1(16384), TPB, 0, stream>>>(encW + i * 65536, encWbar + i * 16384, 4, 16384);
        k_avg_head<<<g1(16384), TPB, 0, stream>>>(encWe + i * 65536, encWebar + i * 16384, 4, 16384);
        k_avg_head<<<g1(128), TPB, 0, stream>>>(encb + i * 512, encbbar + i * 128, 4, 128);
    }
    k_avg_head<<<g1(8192), TPB, 0, stream>>>(dW, decWbar, 4, 8192);
    k_avg_head<<<g1(4096), TPB, 0, stream>>>(dWe, decWebar, 4, 4096);
    k_avg_head<<<g1(64), TPB, 0, stream>>>(db, decbbar, 4, 64);

    // ---- Bottleneck_Construct ----
    gemm(x, nullptr, Wd, bd, Hb, 32768, 64, 64, 0);
    k_pool_conv<<<g1(8 * 1024 * 64), TPB, 0, stream>>>(Hb, K0, Kb0, S1, 8, 1024, 64);
    k_pool_conv<<<g1(8 * 256 * 64), TPB, 0, stream>>>(S1, K1, Kb1, S2, 8, 256, 64);

    // ---- per-scale encoder ----
    const int SEQv[3] = {4096, 1024, 256};
    const int STRD[3] = {1, 4, 16};
    float* SCL[3] = {Hb, S1, S2};
    float* STp[3] = {ST0, ST1, ST2};
    for (int i = 0; i < 3; ++i) {
        int N = SEQv[i], E = N / 4, nnz = 3 * N, rows = 8 * N;
        const int* node = hgp[i];
        const int* edge = hgp[i] + nnz;
        gemm(SCL[i], nullptr, Wu, bu, SE, rows, 64, 64, 0);
        k_layernorm<<<g1(rows), TPB, 0, stream>>>(SE, lng, lnb, SE, rows, 64);
        k_build_e<<<g1((long long)rows * 128), TPB, 0, stream>>>(SE, x, Eb, 8, N, 64, STRD[i], 4096);
        k_fill4<<<g1(E * 8 * 32), TPB, 0, stream>>>((float4*)EF, E * 8 * 32);
        k_fill4<<<g1(E * 8 * 32), TPB, 0, stream>>>((float4*)HE, E * 8 * 32);
        k_fill4<<<g1(E / 4), TPB, 0, stream>>>((float4*)DEG, E / 4);
        k_edeg<<<g1(nnz), TPB, 0, stream>>>(edge, nnz, DEG);
        k_scatter<<<g1((long long)nnz * 8 * 128), TPB, 0, stream>>>(EF, Eb, node, edge, nnz, 8, 128, N);
        gemm(Eb, nullptr, encWbar + i * 16384, nullptr, Yb, rows, 128, 128, 0);
        k_scatter<<<g1((long long)nnz * 8 * 128), TPB, 0, stream>>>(HE, Yb, node, edge, nnz, 8, 128, N);
        gemm(EF, nullptr, encWebar + i * 16384, nullptr, EE, E * 8, 128, 128, 0);
        k_msg4<<<g1(E * 8 * 32), TPB, 0, stream>>>((float4*)HE, (const float4*)EE, DEG, E, 8, 128);
        k_gather4<<<g1((long long)rows * 32), TPB, 0, stream>>>(HE, edge, encbbar + i * 128, Eb, MH, 8, N, 128);
        k_movavg4<<<g1((long long)rows * 32), TPB, 0, stream>>>(MH, TR, 8, N, 128);
        gemm(Eb, nullptr, fWg + i * 16384, fbg + i * 128, TA, rows, 128, 128, 0);
        gemm(MH, TR, fWs + i * 16384, fbs + i * 128, TB, rows, 128, 128, 1);   // sea = mh - trend
        gemm(TR, nullptr, fWt + i * 16384, fbt + i * 128, TC, rows, 128, 128, 0);
        k_stcomb4<<<g1((long long)rows * 32), TPB, 0, stream>>>((const float4*)TA, (const float4*)TB,
                                                                (const float4*)TC, (float4*)STp[i],
                                                                rows * 32, 0);
    }

    // ---- MSFusion ----
    k_upsum4<<<g1(8 * 4096 * 32), TPB, 0, stream>>>(ST0, ST1, ST2, 8, 4096, 128);
    gemm(ST0, nullptr, msW, msb, SE, 32768, 128, 64, 0);    // fused_logits -> SE

    // ---- decoder ----
    const int* dnode = fhg;
    const int* dedge = fhg + 12288;
    k_fill4<<<g1(1024 * 8 * 16), TPB, 0, stream>>>((float4*)EF, 1024 * 8 * 16);
    k_fill4<<<g1(1024 * 8 * 16), TPB, 0, stream>>>((float4*)HE, 1024 * 8 * 16);
    k_fill4<<<g1(256), TPB, 0, stream>>>((float4*)DEG, 256);
    k_edeg<<<g1(12288), TPB, 0, stream>>>(dedge, 12288, DEG);
    k_scatter<<<g1(12288LL * 8 * 64), TPB, 0, stream>>>(EF, x, dnode, dedge, 12288, 8, 64, 4096);
    k_movavg4<<<g1(32768LL * 16), TPB, 0, stream>>>(x, TR, 8, 4096, 64);       // trend1
    k_dec_inp<<<g1(32768LL * 128), TPB, 0, stream>>>(x, TR, SE, Eb, 8, 4096, 64);
    gemm(Eb, nullptr, decWbar, nullptr, Yb, 32768, 128, 64, 0);
    k_scatter<<<g1(12288LL * 8 * 64), TPB, 0, stream>>>(HE, Yb, dnode, dedge, 12288, 8, 64, 4096);
    gemm(EF, nullptr, decWebar, nullptr, EE, 8192, 64, 64, 0);
    k_msg4<<<g1(1024 * 8 * 16), TPB, 0, stream>>>((float4*)HE, (const float4*)EE, DEG, 1024, 8, 64);
    k_gather4<<<g1(32768LL * 16), TPB, 0, stream>>>(HE, dedge, decbbar, nullptr, MH, 8, 4096, 64);
    k_movavg4<<<g1(32768LL * 16), TPB, 0, stream>>>(MH, TA, 8, 4096, 64);      // trend2
    gemm(x, nullptr, dfWg, dfbg, TB, 32768, 64, 64, 0);
    gemm(MH, TA, dfWs, dfbs, TC, 32768, 64, 64, 1);                            // sea2 = mh - trend2
    gemm(TR, TA, dfWt, dfbt, Yb, 32768, 64, 64, 2);                            // trend1 + trend2
    k_stcomb4<<<g1(32768LL * 16), TPB, 0, stream>>>((const float4*)TB, (const float4*)TC,
                                                    (const float4*)Yb, (float4*)d_out,
                                                    32768 * 16, 1);
}